// AttentionLayer_28793460752967
// MI455X (gfx1250) — compile-verified
//
#include <hip/hip_runtime.h>

#define NB 128
#define NL 96
#define ND 512
#define NH 8
#define NE 64
#define NFLAT 49152   // L*D = L*H*E
#define NCOL 64       // ring rank^2

typedef __attribute__((ext_vector_type(2))) float v2f;
typedef __attribute__((ext_vector_type(8))) float v8f;

__device__ __forceinline__ v8f wmma4(v2f a, v2f b, v8f c) {
  // D = A(16x4) * B(4x16) + C, fp32
  return __builtin_amdgcn_wmma_f32_16x16x4_f32(false, a, false, b, (short)0, c,
                                               false, false);
}

// ---------------------------------------------------------------------------
// Builders: collapse the tensor-ring cores into dense [NFLAT, 64] factors.
// ---------------------------------------------------------------------------

// Chain cores 0..5 (modes 2,2,2,2,2,3): U6[x=8][combo=96][k=8]
__global__ void chain6_kernel(const float* __restrict__ c0, const float* __restrict__ c1,
                              const float* __restrict__ c2, const float* __restrict__ c3,
                              const float* __restrict__ c4, const float* __restrict__ c5,
                              float* __restrict__ U6) {
  int t = blockIdx.x * blockDim.x + threadIdx.x;
  if (t >= 96) return;
  int m6 = t % 3;
  int r  = t / 3;
  int m5 = r & 1, m4 = (r >> 1) & 1, m3 = (r >> 2) & 1, m2 = (r >> 3) & 1, m1 = (r >> 4) & 1;
  float M[8][8], Tm[8][8];
  for (int a = 0; a < 8; a++)
    for (int k = 0; k < 8; k++) M[a][k] = c0[(a * 2 + m1) * 8 + k];
  const float* cs[5] = {c1, c2, c3, c4, c5};
  int ms[5] = {m2, m3, m4, m5, m6};
  int md[5] = {2, 2, 2, 2, 3};
  for (int s = 0; s < 5; s++) {
    const float* c = cs[s];
    for (int a = 0; a < 8; a++)
      for (int j = 0; j < 8; j++) {
        float acc = 0.f;
        for (int k = 0; k < 8; k++) acc += M[a][k] * c[(k * md[s] + ms[s]) * 8 + j];
        Tm[a][j] = acc;
      }
    for (int a = 0; a < 8; a++)
      for (int j = 0; j < 8; j++) M[a][j] = Tm[a][j];
  }
  for (int a = 0; a < 8; a++)
    for (int k = 0; k < 8; k++) U6[(a * 96 + t) * 8 + k] = M[a][k];
}

// Expand with last core (8,512,8): W[n=combo*512+m7][col], col = swap ? y*8+x : x*8+y
__global__ void expand7_kernel(const float* __restrict__ U6, const float* __restrict__ c6,
                               float* __restrict__ W, int swap) {
  int n = blockIdx.x * blockDim.x + threadIdx.x;
  if (n >= NFLAT) return;
  int combo = n >> 9, m7 = n & 511;
  float u[8][8];
  for (int x = 0; x < 8; x++)
    for (int k = 0; k < 8; k++) u[x][k] = U6[(x * 96 + combo) * 8 + k];
  float* wrow = W + (size_t)n * NCOL;
  for (int x = 0; x < 8; x++)
    for (int y = 0; y < 8; y++) {
      float acc = 0.f;
      for (int k = 0; k < 8; k++) acc += u[x][k] * c6[(k * 512 + m7) * 8 + y];
      wrow[swap ? (y * 8 + x) : (x * 8 + y)] = acc;
    }
}

// 2-core fuse: A0 (8,96,8), A1 (8,512,8): W[n1*512+n2][col] = sum_p A0[x,n1,p]A1[p,n2,y]
__global__ void fuse2_kernel(const float* __restrict__ A0, const float* __restrict__ A1,
                             float* __restrict__ W, int swap) {
  int n = blockIdx.x * blockDim.x + threadIdx.x;
  if (n >= NFLAT) return;
  int n1 = n >> 9, n2 = n & 511;
  float a0[8][8];
  for (int x = 0; x < 8; x++)
    for (int p = 0; p < 8; p++) a0[x][p] = A0[(x * 96 + n1) * 8 + p];
  float* wrow = W + (size_t)n * NCOL;
  for (int x = 0; x < 8; x++)
    for (int y = 0; y < 8; y++) {
      float acc = 0.f;
      for (int p = 0; p < 8; p++) acc += a0[x][p] * A1[(p * 512 + n2) * 8 + y];
      wrow[swap ? (y * 8 + x) : (x * 8 + y)] = acc;
    }
}

// ---------------------------------------------------------------------------
// GEMM stage 1: X[128, NFLAT] @ W[NFLAT, 64] -> partial[96][128][64] (split-K)
// grid 384 x 256; wave gw: ksub = gw%96, tile = gw/96 (mt 0..7, nt 0..3)
// ---------------------------------------------------------------------------
__global__ __launch_bounds__(256) void gemm1_partial(const float* __restrict__ X,
                                                     const float* __restrict__ W,
                                                     float* __restrict__ partial) {
  int w = threadIdx.x >> 5, lane = threadIdx.x & 31;
  int gw = blockIdx.x * 8 + w;
  int ksub = gw % 96, tile = gw / 96;
  int mt = tile >> 2, nt = tile & 3;
  int lm = lane & 15, kh = (lane >> 4) << 1;
  const float* Ar = X + (size_t)(mt * 16 + lm) * NFLAT + (size_t)ksub * 512;
  const float* Bp = W + (size_t)ksub * 512 * NCOL + nt * 16 + lm;
  v8f acc = {};
  for (int kk = 0; kk < 512; kk += 4) {
    v2f a, b;
    a.x = Ar[kk + kh];
    a.y = Ar[kk + kh + 1];
    b.x = Bp[(size_t)(kk + kh) * NCOL];
    b.y = Bp[(size_t)(kk + kh + 1) * NCOL];
    acc = wmma4(a, b, acc);
  }
  float* out = partial + (size_t)ksub * (128 * NCOL) + (size_t)(mt * 16) * NCOL + nt * 16;
  int rb = (lane >> 4) * 8;
  for (int r = 0; r < 8; r++) out[(size_t)(rb + r) * NCOL + lm] = acc[r];
}

__global__ void reduceK_kernel(const float* __restrict__ partial, float* __restrict__ T) {
  int i = blockIdx.x * blockDim.x + threadIdx.x;  // 8192
  float acc = 0.f;
  for (int s = 0; s < 96; s++) acc += partial[(size_t)s * 8192 + i];
  T[i] = acc;
}

// GEMM stage 2: T[128,64] @ Wout[NFLAT,64]^T -> Y[128, NFLAT]
// grid 3072 x 256; wave gw: mt = gw&7, nt = gw>>3 (0..3071)
__global__ __launch_bounds__(256) void gemm2_kernel(const float* __restrict__ T,
                                                    const float* __restrict__ Wout,
                                                    float* __restrict__ Y) {
  int w = threadIdx.x >> 5, lane = threadIdx.x & 31;
  int gw = blockIdx.x * 8 + w;
  int mt = gw & 7, nt = gw >> 3;
  int lm = lane & 15, kh = (lane >> 4) << 1;
  const float* Tr = T + (size_t)(mt * 16 + lm) * NCOL;
  const float* Wr = Wout + (size_t)(nt * 16 + lm) * NCOL;
  v8f acc = {};
  for (int kk = 0; kk < 64; kk += 4) {
    v2f a, b;
    a.x = Tr[kk + kh];
    a.y = Tr[kk + kh + 1];
    b.x = Wr[kk + kh];
    b.y = Wr[kk + kh + 1];
    acc = wmma4(a, b, acc);
  }
  float* out = Y + (size_t)(mt * 16) * NFLAT + nt * 16;
  int rb = (lane >> 4) * 8;
  for (int r = 0; r < 8; r++) out[(size_t)(rb + r) * NFLAT + lm] = acc[r];
}

// scores[bh][l][s] = (Q_bh K_bh^T)/8 ; q,k layout [B, L, H*E]
// grid 4608 x 256; wave gw: tile = gw%36 (mt,nt in 6x6), bh = gw/36
__global__ __launch_bounds__(256) void scores_kernel(const float* __restrict__ q,
                                                     const float* __restrict__ k,
                                                     float* __restrict__ A) {
  int w = threadIdx.x >> 5, lane = threadIdx.x & 31;
  int gw = blockIdx.x * 8 + w;
  int tile = gw % 36, bh = gw / 36;
  int mt = tile / 6, nt = tile % 6;
  int b = bh >> 3, h = bh & 7;
  int lm = lane & 15, kh = (lane >> 4) << 1;
  const float* Qr = q + ((size_t)b * NL + mt * 16 + lm) * ND + h * NE;
  const float* Kr = k + ((size_t)b * NL + nt * 16 + lm) * ND + h * NE;
  v8f acc = {};
  for (int kk = 0; kk < 64; kk += 4) {
    v2f a, bb;
    a.x = Qr[kk + kh];
    a.y = Qr[kk + kh + 1];
    bb.x = Kr[kk + kh];
    bb.y = Kr[kk + kh + 1];
    acc = wmma4(a, bb, acc);
  }
  float* out = A + ((size_t)bh * NL + mt * 16) * NL + nt * 16;
  int rb = (lane >> 4) * 8;
  for (int r = 0; r < 8; r++) out[(size_t)(rb + r) * NL + lm] = acc[r] * 0.125f;
}

// in-place masked softmax over S=96; one wave per row, 3 elems/lane
__global__ __launch_bounds__(256) void softmax_kernel(float* __restrict__ A,
                                                      const unsigned char* __restrict__ mask) {
  int w = threadIdx.x >> 5, lane = threadIdx.x & 31;
  size_t row = (size_t)blockIdx.x * 8 + w;  // B*H*L rows
  int l = (int)(row % NL);
  float* p = A + row * NL;
  float v0 = mask[l * NL + lane]      ? -__builtin_inff() : p[lane];
  float v1 = mask[l * NL + lane + 32] ? -__builtin_inff() : p[lane + 32];
  float v2 = mask[l * NL + lane + 64] ? -__builtin_inff() : p[lane + 64];
  float m = fmaxf(v0, fmaxf(v1, v2));
  for (int off = 16; off > 0; off >>= 1) m = fmaxf(m, __shfl_xor(m, off, 32));
  float e0 = __expf(v0 - m), e1 = __expf(v1 - m), e2 = __expf(v2 - m);
  float s = e0 + e1 + e2;
  for (int off = 16; off > 0; off >>= 1) s += __shfl_xor(s, off, 32);
  float inv = 1.0f / s;
  p[lane] = e0 * inv;
  p[lane + 32] = e1 * inv;
  p[lane + 64] = e2 * inv;
}

// ctx[b,l,h,e] = sum_s A[bh,l,s] v[b,s,h,e]
// grid 3072 x 256; wave gw: tile = gw%24 (mt 0..5, nt 0..3), bh = gw/24
__global__ __launch_bounds__(256) void av_kernel(const float* __restrict__ A,
                                                 const float* __restrict__ v,
                                                 float* __restrict__ ctx) {
  int w = threadIdx.x >> 5, lane = threadIdx.x & 31;
  int gw = blockIdx.x * 8 + w;
  int tile = gw % 24, bh = gw / 24;
  int mt = tile >> 2, nt = tile & 3;
  int b = bh >> 3, h = bh & 7;
  int lm = lane & 15, kh = (lane >> 4) << 1;
  const float* Ar = A + ((size_t)bh * NL + mt * 16 + lm) * NL;
  const float* Vr = v + ((size_t)b * NL) * ND + h * NE + nt * 16 + lm;
  v8f acc = {};
  for (int kk = 0; kk < 96; kk += 4) {
    v2f a, bb;
    a.x = Ar[kk + kh];
    a.y = Ar[kk + kh + 1];
    bb.x = Vr[(size_t)(kk + kh) * ND];
    bb.y = Vr[(size_t)(kk + kh + 1) * ND];
    acc = wmma4(a, bb, acc);
  }
  float* out = ctx + ((size_t)b * NL + mt * 16) * ND + h * NE + nt * 16;
  int rb = (lane >> 4) * 8;
  for (int r = 0; r < 8; r++) out[(size_t)(rb + r) * ND + lm] = acc[r];
}

// ---------------------------------------------------------------------------

extern "C" void kernel_launch(void* const* d_in, const int* in_sizes, int n_in,
                              void* d_out, int out_size, void* d_ws, size_t ws_size,
                              hipStream_t stream) {
  const float* queries = (const float*)d_in[0];
  const float* keys    = (const float*)d_in[1];
  const float* values  = (const float*)d_in[2];
  const unsigned char* mask = (const unsigned char*)d_in[3];
  // q_cores: in 4..10, out 11..12 ; k: 13..19, 20..21 ; v: 22..28, 29..30
  // o_cores: in 31..32 (modes 96,512), out 33..39 (modes 2,2,2,2,2,3,512)
  const float* qin[7];  for (int i = 0; i < 7; i++) qin[i]  = (const float*)d_in[4 + i];
  const float* qout[2]; for (int i = 0; i < 2; i++) qout[i] = (const float*)d_in[11 + i];
  const float* kin[7];  for (int i = 0; i < 7; i++) kin[i]  = (const float*)d_in[13 + i];
  const float* kout[2]; for (int i = 0; i < 2; i++) kout[i] = (const float*)d_in[20 + i];
  const float* vin[7];  for (int i = 0; i < 7; i++) vin[i]  = (const float*)d_in[22 + i];
  const float* vout[2]; for (int i = 0; i < 2; i++) vout[i] = (const float*)d_in[29 + i];
  const float* oin[2];  for (int i = 0; i < 2; i++) oin[i]  = (const float*)d_in[31 + i];
  const float* oout[7]; for (int i = 0; i < 7; i++) oout[i] = (const float*)d_in[33 + i];

  float* ws = (float*)d_ws;
  size_t off = 0;
  const size_t WSZ = (size_t)NFLAT * NCOL;
  float* Wq_in  = ws + off; off += WSZ;
  float* Wk_in  = ws + off; off += WSZ;
  float* Wv_in  = ws + off; off += WSZ;
  float* Wq_out = ws + off; off += WSZ;
  float* Wk_out = ws + off; off += WSZ;
  float* Wv_out = ws + off; off += WSZ;
  float* Wo_in  = ws + off; off += WSZ;
  float* Wo_out = ws + off; off += WSZ;
  float* U6q = ws + off; off += 6144;
  float* U6k = ws + off; off += 6144;
  float* U6v = ws + off; off += 6144;
  float* U6o = ws + off; off += 6144;
  float* Tq = ws + off; off += 8192;
  float* Tk = ws + off; off += 8192;
  float* Tv = ws + off; off += 8192;
  float* To = ws + off; off += 8192;
  float* partial = ws + off; off += (size_t)96 * 8192;
  float* qbuf = ws + off; off += (size_t)NB * NFLAT;
  float* kbuf = ws + off; off += (size_t)NB * NFLAT;
  float* vbuf = ws + off; off += (size_t)NB * NFLAT;
  float* ctx  = ws + off; off += (size_t)NB * NFLAT;

  float* out0 = (float*)d_out;                       // [B, L, D]
  float* Aout = out0 + (size_t)NB * NL * ND;         // [B, H, L, S]

  // --- build dense rank-64 factors from the tensor-ring cores ---
  chain6_kernel<<<1, 128, 0, stream>>>(qin[0], qin[1], qin[2], qin[3], qin[4], qin[5], U6q);
  chain6_kernel<<<1, 128, 0, stream>>>(kin[0], kin[1], kin[2], kin[3], kin[4], kin[5], U6k);
  chain6_kernel<<<1, 128, 0, stream>>>(vin[0], vin[1], vin[2], vin[3], vin[4], vin[5], U6v);
  chain6_kernel<<<1, 128, 0, stream>>>(oout[0], oout[1], oout[2], oout[3], oout[4], oout[5], U6o);
  expand7_kernel<<<192, 256, 0, stream>>>(U6q, qin[6], Wq_in, 0);
  expand7_kernel<<<192, 256, 0, stream>>>(U6k, kin[6], Wk_in, 0);
  expand7_kernel<<<192, 256, 0, stream>>>(U6v, vin[6], Wv_in, 0);
  expand7_kernel<<<192, 256, 0, stream>>>(U6o, oout[6], Wo_out, 1);
  fuse2_kernel<<<192, 256, 0, stream>>>(qout[0], qout[1], Wq_out, 1);
  fuse2_kernel<<<192, 256, 0, stream>>>(kout[0], kout[1], Wk_out, 1);
  fuse2_kernel<<<192, 256, 0, stream>>>(vout[0], vout[1], Wv_out, 1);
  fuse2_kernel<<<192, 256, 0, stream>>>(oin[0], oin[1], Wo_in, 0);

  // --- q/k/v projections: x @ W_in -> T[128,64]; T @ W_out^T -> [128, NFLAT]
  gemm1_partial<<<384, 256, 0, stream>>>(queries, Wq_in, partial);
  reduceK_kernel<<<32, 256, 0, stream>>>(partial, Tq);
  gemm2_kernel<<<3072, 256, 0, stream>>>(Tq, Wq_out, qbuf);
  gemm1_partial<<<384, 256, 0, stream>>>(keys, Wk_in, partial);
  reduceK_kernel<<<32, 256, 0, stream>>>(partial, Tk);
  gemm2_kernel<<<3072, 256, 0, stream>>>(Tk, Wk_out, kbuf);
  gemm1_partial<<<384, 256, 0, stream>>>(values, Wv_in, partial);
  reduceK_kernel<<<32, 256, 0, stream>>>(partial, Tv);
  gemm2_kernel<<<3072, 256, 0, stream>>>(Tv, Wv_out, vbuf);

  // --- attention ---
  scores_kernel<<<4608, 256, 0, stream>>>(qbuf, kbuf, Aout);
  softmax_kernel<<<12288, 256, 0, stream>>>(Aout, mask);
  av_kernel<<<3072, 256, 0, stream>>>(Aout, vbuf, ctx);

  // --- output projection ---
  gemm1_partial<<<384, 256, 0, stream>>>(ctx, Wo_in, partial);
  reduceK_kernel<<<32, 256, 0, stream>>>(partial, To);
  gemm2_kernel<<<3072, 256, 0, stream>>>(To, Wo_out, out0);
}